// Router_44890998177816
// MI455X (gfx1250) — compile-verified
//
#include <hip/hip_runtime.h>
#include <hip/hip_bf16.h>
#include <math.h>

// ---------------------------------------------------------------------------
// MoE router for MI455X (gfx1250, wave32).
//   logits = x[16384,4096] @ W[64,4096]^T + b[64]      (fp32 WMMA 16x16x4)
//   probs  = softmax(logits); top-2; weights = softmax(top2 probs)
//   mask   = one_hot(idx).transpose(2,1,0)  -> [E, K, N]
// ---------------------------------------------------------------------------

#define NUM_EXPERT 64
#define TOP_K      2
#define N_TOKENS   16384
#define H_DIM      4096

typedef __attribute__((ext_vector_type(2))) float v2f;
typedef __attribute__((ext_vector_type(8))) float v8f;

// One wave computes a 16-token x 64-expert strip of logits.
// A (16x4 f32): lane = token row (M = lane&15); lanes 0-15 carry K slots 0,1;
// lanes 16-31 carry K slots 2,3.  We assign global k..k+3 to the low half and
// k+4..k+7 to the high half (a K-permutation, harmless for a dot product), so
// every lane issues one contiguous B128 load per operand per 8-k step.
__global__ __launch_bounds__(256, 2) void router_gemm_kernel(
    const float* __restrict__ x, const float* __restrict__ W,
    const float* __restrict__ bias, float* __restrict__ logits)
{
    const int lane = threadIdx.x & 31;
    const int wave = threadIdx.x >> 5;
    const int tile = blockIdx.x * 8 + wave;   // 16-token tile, 0..1023
    const int m    = lane & 15;               // token row within tile / expert col
    const int half = lane >> 4;               // K-slot half select
    const int row0 = tile * 16;

    const float* xr = x + (size_t)(row0 + m) * H_DIM + half * 4;
    const float* w0 = W + (size_t)(m +  0) * H_DIM + half * 4;
    const float* w1 = W + (size_t)(m + 16) * H_DIM + half * 4;
    const float* w2 = W + (size_t)(m + 32) * H_DIM + half * 4;
    const float* w3 = W + (size_t)(m + 48) * H_DIM + half * 4;

    v8f acc0 = {}, acc1 = {}, acc2 = {}, acc3 = {};

#pragma unroll 2
    for (int k = 0; k < H_DIM; k += 8) {
        float4 a  = *(const float4*)(xr + k);
        float4 p0 = *(const float4*)(w0 + k);
        float4 p1 = *(const float4*)(w1 + k);
        float4 p2 = *(const float4*)(w2 + k);
        float4 p3 = *(const float4*)(w3 + k);

        v2f alo = {a.x, a.y};
        v2f ahi = {a.z, a.w};
        v2f q;

        q = (v2f){p0.x, p0.y};
        acc0 = __builtin_amdgcn_wmma_f32_16x16x4_f32(false, alo, false, q, (short)0, acc0, false, false);
        q = (v2f){p1.x, p1.y};
        acc1 = __builtin_amdgcn_wmma_f32_16x16x4_f32(false, alo, false, q, (short)0, acc1, false, false);
        q = (v2f){p2.x, p2.y};
        acc2 = __builtin_amdgcn_wmma_f32_16x16x4_f32(false, alo, false, q, (short)0, acc2, false, false);
        q = (v2f){p3.x, p3.y};
        acc3 = __builtin_amdgcn_wmma_f32_16x16x4_f32(false, alo, false, q, (short)0, acc3, false, false);

        q = (v2f){p0.z, p0.w};
        acc0 = __builtin_amdgcn_wmma_f32_16x16x4_f32(false, ahi, false, q, (short)0, acc0, false, false);
        q = (v2f){p1.z, p1.w};
        acc1 = __builtin_amdgcn_wmma_f32_16x16x4_f32(false, ahi, false, q, (short)0, acc1, false, false);
        q = (v2f){p2.z, p2.w};
        acc2 = __builtin_amdgcn_wmma_f32_16x16x4_f32(false, ahi, false, q, (short)0, acc2, false, false);
        q = (v2f){p3.z, p3.w};
        acc3 = __builtin_amdgcn_wmma_f32_16x16x4_f32(false, ahi, false, q, (short)0, acc3, false, false);
    }

    // C layout: VGPR v holds token row (row0 + half*8 + v), lane&15 = expert.
    const float b0 = bias[m +  0];
    const float b1 = bias[m + 16];
    const float b2 = bias[m + 32];
    const float b3 = bias[m + 48];

    float* out = logits + (size_t)(row0 + half * 8) * NUM_EXPERT + m;
#pragma unroll
    for (int v = 0; v < 8; ++v) {
        out[(size_t)v * NUM_EXPERT +  0] = acc0[v] + b0;
        out[(size_t)v * NUM_EXPERT + 16] = acc1[v] + b1;
        out[(size_t)v * NUM_EXPERT + 32] = acc2[v] + b2;
        out[(size_t)v * NUM_EXPERT + 48] = acc3[v] + b3;
    }
}

// One thread per token: top-2 (lowest-index tie order, matching lax.top_k),
// softmax denominator, then the reference's second softmax over the two
// probabilities: w1 = 1/(1+exp(p2-p1)), w2 = 1-w1 (p2 <= p1, so stable).
__global__ __launch_bounds__(256) void router_topk_kernel(
    const float* __restrict__ logits, float* __restrict__ weights,
    float* __restrict__ sel)
{
    const int t = blockIdx.x * 256 + threadIdx.x;
    const float* L = logits + (size_t)t * NUM_EXPERT;

    float v1 = -INFINITY, v2 = -INFINITY;
    int   i1 = 0, i2 = 0;
#pragma unroll 8
    for (int e = 0; e < NUM_EXPERT; ++e) {
        const float l = L[e];
        const bool gt1 = l > v1;
        const bool gt2 = l > v2;
        v2 = gt1 ? v1 : (gt2 ? l : v2);
        i2 = gt1 ? i1 : (gt2 ? e : i2);
        v1 = gt1 ? l  : v1;
        i1 = gt1 ? e  : i1;
    }

    float sum = 0.f;
#pragma unroll 8
    for (int e = 0; e < NUM_EXPERT; ++e)
        sum += __expf(L[e] - v1);

    const float inv = 1.f / sum;
    const float p1  = inv;                    // exp(v1-v1)/sum
    const float p2  = __expf(v2 - v1) * inv;

    const float e21 = __expf(p2 - p1);
    const float w1  = 1.f / (1.f + e21);
    const float w2  = e21 * w1;

    weights[t * TOP_K + 0] = w1;
    weights[t * TOP_K + 1] = w2;
    sel[t * TOP_K + 0] = (float)i1;
    sel[t * TOP_K + 1] = (float)i2;
}

// expert_mask [E, K, N] (int one-hot, stored as 0.0/1.0 floats), fully
// coalesced: flat idx = ((e*2 + k) << 14) + n  since N = 2^14.
__global__ __launch_bounds__(256) void router_mask_kernel(
    const float* __restrict__ sel, float* __restrict__ mask)
{
    const int idx = blockIdx.x * 256 + threadIdx.x;
    const int n  = idx & (N_TOKENS - 1);
    const int ek = idx >> 14;       // e*TOP_K + k
    const int k  = ek & 1;
    const int e  = ek >> 1;
    mask[idx] = (sel[n * TOP_K + k] == (float)e) ? 1.0f : 0.0f;
}

extern "C" void kernel_launch(void* const* d_in, const int* in_sizes, int n_in,
                              void* d_out, int out_size, void* d_ws, size_t ws_size,
                              hipStream_t stream)
{
    const float* x = (const float*)d_in[0];
    const float* W = (const float*)d_in[1];
    const float* b = (const float*)d_in[2];

    float* logits  = (float*)d_out;                                   // [N, E]
    float* weights = logits  + (size_t)N_TOKENS * NUM_EXPERT;         // [N, K]
    float* sel     = weights + (size_t)N_TOKENS * TOP_K;              // [N, K]
    float* mask    = sel     + (size_t)N_TOKENS * TOP_K;              // [E, K, N]

    // 1024 token-tiles of 16, 8 waves (tiles) per 256-thread block.
    router_gemm_kernel<<<(N_TOKENS / 16) / 8, 256, 0, stream>>>(x, W, b, logits);
    router_topk_kernel<<<N_TOKENS / 256, 256, 0, stream>>>(logits, weights, sel);
    router_mask_kernel<<<(NUM_EXPERT * TOP_K * N_TOKENS) / 256, 256, 0, stream>>>(sel, mask);
}